// AlphaCell_72980084294308
// MI455X (gfx1250) — compile-verified
//
#include <hip/hip_runtime.h>
#include <stdint.h>

// Alpha filter scan: B=64, T=1024, K=512, fp32.
// Memory-bound (268 MB @ 23.3 TB/s ~ 11.5us floor). CDNA5 path: TDM
// tensor_load_to_lds double-buffered input staging + s_wait_tensorcnt,
// wave32-sized blocks, nontemporal output stores.

#define DT_S   0.001f
#define BATCH_ 64
#define SEQ_   1024
#define SIZE_  512
#define KB     256              // features handled per block (1 per lane)
#define CHUNK  16               // timesteps per TDM tile (tile = 16 KB)
#define NCHUNK (SEQ_ / CHUNK)   // 64

typedef unsigned int v4u __attribute__((ext_vector_type(4)));
typedef int          v8i __attribute__((ext_vector_type(8)));
typedef int          v4i __attribute__((ext_vector_type(4)));

// Issue one TDM load: 2-D tile (dim0 = KB floats contiguous, dim1 = CHUNK rows,
// row stride = SIZE_ floats) from global `gsrc` into LDS byte-offset `lds_off`.
__device__ __forceinline__ void tdm_issue(const float* gsrc, uint32_t lds_off) {
    uint64_t ga = (uint64_t)(uintptr_t)gsrc;
    v4u g0;
    g0.x = 1u;                                               // count=1, user mode
    g0.y = lds_off;                                          // lds_addr (bytes)
    g0.z = (uint32_t)(ga & 0xFFFFFFFFu);                     // global_addr[31:0]
    g0.w = (uint32_t)((ga >> 32) & 0x1FFFFFFu) | (2u << 30); // addr[56:32] | type=2
    v8i g1;
    g1[0] = (int)(2u << 16);                                 // data_size=4B, mask=0
    g1[1] = (int)((uint32_t)KB << 16);                       // tensor_dim0[15:0]=256
    g1[2] = (int)((uint32_t)CHUNK << 16);                    // dim0 hi=0 | tensor_dim1=16
    g1[3] = (int)((uint32_t)KB << 16);                       // dim1 hi=0 | tile_dim0=256
    g1[4] = (int)CHUNK;                                      // tile_dim1=16, tile_dim2=0
    g1[5] = (int)SIZE_;                                      // tensor_dim0_stride=512
    g1[6] = 0;
    g1[7] = 0;
    v4i z4 = {0, 0, 0, 0};                                   // groups 2/3: 2-D tensor
    v8i z8 = {0, 0, 0, 0, 0, 0, 0, 0};                       // extra group (clang-23 form)
    __builtin_amdgcn_tensor_load_to_lds(g0, g1, z4, z4, z8, 0);
}

__global__ __launch_bounds__(KB) void AlphaCell_scan_kernel(
        const float* __restrict__ in,          // [B, T, K]
        const float* __restrict__ init_level,  // [K]
        const float* __restrict__ tau,         // [K]
        float* __restrict__ out) {             // [B, T, K]
    __shared__ float lbuf[2][CHUNK * KB];      // 32 KB double buffer

    const int kloc = threadIdx.x;              // 0..255
    const int kblk = blockIdx.x;               // 0..1
    const int b    = blockIdx.y;               // 0..63
    const int k    = kblk * KB + kloc;

    // Per-feature coefficients: A = e*[[1-dt, -dt2],[DT, 1+dt]], B = [e*dt2, 1-e*(1+dt)]
    float tc      = fmaxf(tau[k], 1e-8f);
    float dt_tau  = DT_S / tc;
    float dt_tau2 = dt_tau / tc;
    float e       = expf(-dt_tau);
    float a00 = e * (1.0f - dt_tau);
    float a01 = -e * dt_tau2;
    float a10 = e * DT_S;
    float a11 = e * (1.0f + dt_tau);
    float b0  = e * dt_tau2;
    float b1  = 1.0f - e * (1.0f + dt_tau);

    float s0 = 0.0f;
    float s1 = init_level[k];

    const float* src = in  + (size_t)b * SEQ_ * SIZE_ + (size_t)kblk * KB;
    float*       dst = out + (size_t)b * SEQ_ * SIZE_ + k;

    const uint32_t lds0 = (uint32_t)(uintptr_t)&lbuf[0][0];
    const uint32_t lds1 = (uint32_t)(uintptr_t)&lbuf[1][0];

    // Prologue: prefetch chunks 0 and 1 (TENSORcnt = 2 on wave 0).
    if (threadIdx.x == 0) {
        tdm_issue(src, lds0);
        tdm_issue(src + (size_t)CHUNK * SIZE_, lds1);
    }

    for (int c = 0; c < NCHUNK; ++c) {
        if (threadIdx.x == 0) {
            // TDM ops are in-order per wave: <=1 outstanding ==> chunk c landed.
            if (c + 1 < NCHUNK) __builtin_amdgcn_s_wait_tensorcnt(1);
            else                __builtin_amdgcn_s_wait_tensorcnt(0);
        }
        __syncthreads();                      // chunk c visible to all waves

        const float* lb = lbuf[c & 1];
        float* o = dst + (size_t)c * CHUNK * SIZE_;
        #pragma unroll
        for (int r = 0; r < CHUNK; ++r) {
            float u  = lb[r * KB + kloc];                      // conflict-free ds_load
            float n0 = fmaf(a00, s0, fmaf(a01, s1, b0 * u));
            float n1 = fmaf(a10, s0, fmaf(a11, s1, b1 * u));
            s0 = n0; s1 = n1;
            __builtin_nontemporal_store(s1, o + (size_t)r * SIZE_);  // streamed out
        }
        __syncthreads();                      // everyone done reading buf[c&1]

        if (threadIdx.x == 0 && c + 2 < NCHUNK) {
            tdm_issue(src + (size_t)(c + 2) * CHUNK * SIZE_, (c & 1) ? lds1 : lds0);
        }
    }
}

extern "C" void kernel_launch(void* const* d_in, const int* in_sizes, int n_in,
                              void* d_out, int out_size, void* d_ws, size_t ws_size,
                              hipStream_t stream) {
    const float* in   = (const float*)d_in[0];  // inputs [64,1024,512] f32
    const float* il   = (const float*)d_in[1];  // initial_level [512] f32
    const float* tau  = (const float*)d_in[2];  // tau [512] f32
    float*       outp = (float*)d_out;          // [64,1024,512] f32

    dim3 grid(SIZE_ / KB, BATCH_);              // (2, 64) = 128 blocks
    dim3 block(KB);                             // 256 threads = 8 wave32
    AlphaCell_scan_kernel<<<grid, block, 0, stream>>>(in, il, tau, outp);
}